// AdvancedNODE_55989193671487
// MI455X (gfx1250) — compile-verified
//
#include <hip/hip_runtime.h>
#include <stdint.h>

// Problem constants (fixed by the reference).
#define B_N 16384
#define T_N 256
#define D_N 8
#define F_N 512
#define C_N 16
#define L_N 256

// Tiling.
#define S_TILE 32           // samples per block
#define TG     8            // tree groups (= waves) per block
#define TPG    32           // trees per group
// TDM pad: +1 dword every 256 dwords -> padded LDS row stride = 514 dwords.
#define ROW_STRIDE 514

typedef unsigned int u32x4 __attribute__((ext_vector_type(4)));
typedef int          i32x8 __attribute__((ext_vector_type(8)));
typedef int          i32x4 __attribute__((ext_vector_type(4)));
typedef float        v2f   __attribute__((ext_vector_type(2)));
typedef float        v8f   __attribute__((ext_vector_type(8)));

#if defined(__has_builtin)
#if __has_builtin(__builtin_amdgcn_tensor_load_to_lds)
#define USE_TDM 1
#endif
#if __has_builtin(__builtin_amdgcn_wmma_f32_16x16x4_f32)
#define USE_WMMA 1
#endif
#endif

__device__ __forceinline__ int xidx(int s, int f) {
  // padded layout produced by TDM: pad dword after every 256 dwords
  return s * ROW_STRIDE + f + (f >> 8);
}

// ---------------------------------------------------------------------------
// Kernel 1: fidx[t,d] = argmax_f feature_weights[t,d,f]  (first-max tiebreak)
// One wave32 per row; 8 rows per 256-thread block.
// ---------------------------------------------------------------------------
__global__ __launch_bounds__(256) void argmax_kernel(
    const float* __restrict__ fw, int* __restrict__ fidx) {
  const int row  = blockIdx.x * 8 + (threadIdx.x >> 5);   // 0 .. T_N*D_N-1
  const int lane = threadIdx.x & 31;
  const float* p = fw + (size_t)row * F_N;

  float best = -3.402823466e38f;
  int   bi   = 0;
  for (int f = lane; f < F_N; f += 32) {          // coalesced
    float v = p[f];
    if (v > best) { best = v; bi = f; }           // strict '>' keeps earliest
  }
  // wave32 reduction; on tie prefer lower index (jnp.argmax semantics)
  for (int off = 16; off > 0; off >>= 1) {
    float ov = __shfl_xor(best, off, 32);
    int   oi = __shfl_xor(bi,   off, 32);
    if (ov > best || (ov == best && oi < bi)) { best = ov; bi = oi; }
  }
  if (lane == 0) fidx[row] = bi;
}

// ---------------------------------------------------------------------------
// Kernel 2: forest evaluation.
//   block  -> 32 samples x 256 trees
//   wave g -> trees [g*32, g*32+32), lane = sample
// ---------------------------------------------------------------------------
__global__ __launch_bounds__(256) void forest_kernel(
    const float* __restrict__ x,
    const float* __restrict__ thresholds,
    const float* __restrict__ responses,
    const int*   __restrict__ fidx,
    float*       __restrict__ out) {
  extern __shared__ unsigned char smem[];
  float* lds_x   = (float*)smem;                                   // 32*514 f32
  float* lds_thr = (float*)(smem + S_TILE * ROW_STRIDE * 4);       // 2048 f32
  int*   lds_fx  = (int*)  (smem + S_TILE * ROW_STRIDE * 4 + T_N * D_N * 4);
  float* lds_par = (float*)(smem + S_TILE * ROW_STRIDE * 4 + 2 * T_N * D_N * 4);

  const int tid  = threadIdx.x;
  const int row0 = blockIdx.x * S_TILE;

  // Stage thresholds + fidx (2048 each), coalesced.
#pragma unroll
  for (int i = 0; i < 8; ++i) {
    const int j = i * 256 + tid;
    lds_thr[j] = thresholds[j];
    lds_fx[j]  = fidx[j];
  }

#ifdef USE_TDM
  if (tid == 0) {
    // ---- Tensor DMA descriptor: 2-D tile, 512 x 32 rows of f32,
    //      LDS padding: 1 dword per 256 dwords -> row stride 514 dwords.
    const uint64_t ga    = (uint64_t)(uintptr_t)(x + (size_t)row0 * F_N);
    const uint32_t lbase = (uint32_t)(uintptr_t)lds_x;   // flat low 32 = LDS off

    u32x4 g0;
    g0[0] = 1u;                                           // count=1, no gather
    g0[1] = lbase;                                        // lds_addr
    g0[2] = (uint32_t)ga;                                 // global_addr[31:0]
    g0[3] = ((uint32_t)(ga >> 32) & 0x01FFFFFFu)          // global_addr[56:32]
            | (2u << 30);                                 // type = 2 (image)

    // group1: data_size=4B(2), pad_enable, pad_interval=7(256dw), pad_amt=0(1dw)
    const uint64_t q0 = 0x1D20000ull                      // flags
                      | ((uint64_t)(F_N & 0xFFFF) << 48); // tensor_dim0 lo16
    const uint64_t q1 = ((uint64_t)S_TILE << 16)          // tensor_dim1 @111:80
                      | ((uint64_t)F_N   << 48);          // tile_dim0   @127:112
    const uint64_t q2 = (uint64_t)S_TILE                  // tile_dim1   @143:128
                      | ((uint64_t)F_N   << 32);          // dim0_stride @207:160
    const uint64_t q3 = 0;                                // no dim1 stride (2D)

    i32x8 g1;
    g1[0] = (int)(uint32_t)q0; g1[1] = (int)(uint32_t)(q0 >> 32);
    g1[2] = (int)(uint32_t)q1; g1[3] = (int)(uint32_t)(q1 >> 32);
    g1[4] = (int)(uint32_t)q2; g1[5] = (int)(uint32_t)(q2 >> 32);
    g1[6] = (int)(uint32_t)q3; g1[7] = (int)(uint32_t)(q3 >> 32);

    i32x4 g2 = (i32x4)0;
    i32x4 g3 = (i32x4)0;
    i32x8 g4 = (i32x8)0;     // extra group required by 6-arg toolchain variant

    __builtin_amdgcn_tensor_load_to_lds(g0, g1, g2, g3, g4, 0);
    __builtin_amdgcn_s_wait_tensorcnt(0);
  }
#else
  // Fallback: manual staging into the same padded layout.
  for (int i = tid; i < S_TILE * F_N; i += 256) {
    const int r = i >> 9, c = i & (F_N - 1);
    lds_x[xidx(r, c)] = x[(size_t)(row0 + r) * F_N + c];
  }
#endif
  __syncthreads();

  const int s  = tid & 31;       // sample lane
  const int g  = tid >> 5;       // tree group (wave id)
  const int t0 = g * TPG;

  // Prefetch this group's slice of the response table into L2 (gfx1250 path).
  __builtin_prefetch(responses + ((size_t)t0 * L_N) * C_N + (size_t)s * 64, 0, 1);

  float acc[C_N];
#pragma unroll
  for (int c = 0; c < C_N; ++c) acc[c] = 0.0f;

  for (int t = t0; t < t0 + TPG; ++t) {
    const float* th = &lds_thr[t * D_N];
    const int*   fx = &lds_fx [t * D_N];
    int leaf = 0;
#pragma unroll
    for (int d = 0; d < D_N; ++d) {
      const int   f = fx[d];                 // LDS broadcast (uniform in wave)
      const float v = lds_x[xidx(s, f)];     // bank-conflict-free gather
      leaf = (leaf << 1) | (v > th[d] ? 1 : 0);
    }
    const float4* r =
        (const float4*)(responses + ((size_t)t * L_N + leaf) * C_N);
#pragma unroll
    for (int q = 0; q < 4; ++q) {            // 4x global_load_b128, L2-resident
      const float4 rv = r[q];
      acc[q * 4 + 0] += rv.x;
      acc[q * 4 + 1] += rv.y;
      acc[q * 4 + 2] += rv.z;
      acc[q * 4 + 3] += rv.w;
    }
  }

  // Write partials: P[g][pair], pair = s*16 + c  (512 pairs per block).
#pragma unroll
  for (int c = 0; c < C_N; ++c)
    lds_par[(g * S_TILE + s) * C_N + c] = acc[c];
  __syncthreads();

#ifdef USE_WMMA
  // ---- Cross-group reduction as exact f32 WMMA K-reduction:
  //   D = A(16x4) x B(4x16) + C, B = all-ones -> D[m][n] = sum_k A[m][k].
  //   A[m][k] = P[k (+4 for 2nd wmma)][base + m]; two chained WMMAs sum g=0..7.
  // A layout (ISA 7.12.2, 32-bit A 16x16x4): lane M = lane%16;
  //   lanes 0-15: VGPR0=K0, VGPR1=K1; lanes 16-31: VGPR0=K2, VGPR1=K3.
  // D layout: lane l, VGPR r -> M = r + 8*(l>=16), N = l%16 (all N identical).
  const int lane = tid & 31;
  const int hlf  = lane >> 4;          // which half of the wave
  const int pcol = lane & 15;          // A-matrix row m
  const int gA   = hlf * 2;            // group for VGPR0 of A
  const v2f bones = {1.0f, 1.0f};
  const v8f czero = {0.f, 0.f, 0.f, 0.f, 0.f, 0.f, 0.f, 0.f};
#pragma unroll
  for (int i = 0; i < 4; ++i) {
    const int base = (g * 4 + i) * 16;             // 16 pairs per chain
    v2f a0, a1;
    a0[0] = lds_par[(gA + 0) * (S_TILE * C_N) + base + pcol];   // groups 0..3
    a0[1] = lds_par[(gA + 1) * (S_TILE * C_N) + base + pcol];
    a1[0] = lds_par[(gA + 4) * (S_TILE * C_N) + base + pcol];   // groups 4..7
    a1[1] = lds_par[(gA + 5) * (S_TILE * C_N) + base + pcol];
    v8f d = __builtin_amdgcn_wmma_f32_16x16x4_f32(
        false, a0, false, bones, (short)0, czero, false, false);
    d = __builtin_amdgcn_wmma_f32_16x16x4_f32(
        false, a1, false, bones, (short)0, d, false, false);
    if (pcol < 8) {                                // lanes 0-7 / 16-23 store
      const int m = hlf * 8 + pcol;                // D row held in VGPR pcol
      const int p = base + m;                      // global pair id
      const int ss = p >> 4, cc = p & 15;
      out[((size_t)(row0 + ss)) * C_N + cc] = d[pcol] * (1.0f / T_N);
    }
  }
#else
#pragma unroll
  for (int p = tid; p < S_TILE * C_N; p += 256) {
    const int ss = p >> 4, cc = p & 15;
    float sum = 0.0f;
#pragma unroll
    for (int gg = 0; gg < TG; ++gg)
      sum += lds_par[(gg * S_TILE + ss) * C_N + cc];
    out[((size_t)(row0 + ss)) * C_N + cc] = sum * (1.0f / T_N);
  }
#endif
}

// ---------------------------------------------------------------------------
extern "C" void kernel_launch(void* const* d_in, const int* in_sizes, int n_in,
                              void* d_out, int out_size, void* d_ws,
                              size_t ws_size, hipStream_t stream) {
  const float* x    = (const float*)d_in[0];   // (B, F)
  const float* fw   = (const float*)d_in[1];   // (T, D, F)
  const float* th   = (const float*)d_in[2];   // (T, D)
  const float* resp = (const float*)d_in[3];   // (T, L, C)
  float*       out  = (float*)d_out;           // (B, C)
  int*         fidx = (int*)d_ws;              // (T*D) scratch

  argmax_kernel<<<(T_N * D_N) / 8, 256, 0, stream>>>(fw, fidx);

  const size_t shmem = (size_t)S_TILE * ROW_STRIDE * 4   // x tile (padded)
                     + (size_t)2 * T_N * D_N * 4          // thr + fidx
                     + (size_t)S_TILE * TG * C_N * 4;     // partials
  forest_kernel<<<B_N / S_TILE, 256, shmem, stream>>>(x, th, resp, fidx, out);
}